// GNNModel_32890859553001
// MI455X (gfx1250) — compile-verified
//
#include <hip/hip_runtime.h>

#define DF 128

typedef __attribute__((ext_vector_type(2))) float v2f;
typedef __attribute__((ext_vector_type(8))) float v8f;

// ---------------------------------------------------------------------------
// GEMM: out[n x 128] = (relu_in ? relu(A) : A) @ W,  W is 128x128 row-major.
// Block = 256 threads = 8 waves. Block covers 16 rows; wave w owns cols
// [16w, 16w+16). K tiled by 32 through LDS. Uses V_WMMA_F32_16X16X4_F32.
// ---------------------------------------------------------------------------
__global__ __launch_bounds__(256) void gcn_gemm_wmma(
    const float* __restrict__ A, const float* __restrict__ W,
    float* __restrict__ out, int n, int relu_in)
{
    __shared__ float sA[16][36];    // 16 x 32 chunk of A (padded)
    __shared__ float sW[32][136];   // 32 x 128 chunk of W (stride%32==8 -> halves hit disjoint banks)

    const int tid  = threadIdx.x;
    const int wv   = tid >> 5;      // wave id 0..7  -> column tile
    const int lane = tid & 31;
    const int half = lane >> 4;     // 0: lanes 0-15, 1: lanes 16-31
    const int r    = lane & 15;
    const int row0 = blockIdx.x * 16;

    v8f acc = {};

    for (int kk = 0; kk < DF; kk += 32) {
        // ---- stage A tile: 16 rows x 32 cols (2 floats / thread) ----
        {
            int idx  = tid * 2;            // 0..510
            int ar   = idx >> 5;           // 0..15
            int ac   = idx & 31;           // even
            int grow = row0 + ar;
            grow = grow < n ? grow : n - 1;
            float2 v = *(const float2*)(A + (size_t)grow * DF + kk + ac);
            if (relu_in) { v.x = fmaxf(v.x, 0.f); v.y = fmaxf(v.y, 0.f); }
            sA[ar][ac]     = v.x;
            sA[ar][ac + 1] = v.y;
        }
        // ---- stage W slice: 32 rows x 128 cols (16 floats / thread) ----
        {
            int wr = tid >> 3;             // 0..31
            int wc = (tid & 7) * 16;       // 0..112
            const float4* s4 = (const float4*)(W + (size_t)(kk + wr) * DF + wc);
            float4 w0 = s4[0], w1 = s4[1], w2 = s4[2], w3 = s4[3];
            *(float4*)&sW[wr][wc +  0] = w0;
            *(float4*)&sW[wr][wc +  4] = w1;
            *(float4*)&sW[wr][wc +  8] = w2;
            *(float4*)&sW[wr][wc + 12] = w3;
        }
        __syncthreads();

        #pragma unroll
        for (int k0 = 0; k0 < 32; k0 += 4) {
            // A 16x4 layout: lanes 0-15 K={0,1}, lanes 16-31 K={2,3}
            float2 av = *(const float2*)&sA[r][k0 + 2 * half];
            v2f a; a.x = av.x; a.y = av.y;
            // B 4x16 layout: vgpr0 = rows K={0|2}, vgpr1 = rows K={1|3}
            v2f b;
            b.x = sW[k0 + 2 * half    ][wv * 16 + r];
            b.y = sW[k0 + 2 * half + 1][wv * 16 + r];
            acc = __builtin_amdgcn_wmma_f32_16x16x4_f32(
                false, a, false, b, (short)0, acc, false, false);
        }
        __syncthreads();
    }

    // D 16x16 layout: vgpr i -> row i (lanes 0-15) / row i+8 (lanes 16-31)
    #pragma unroll
    for (int i = 0; i < 8; ++i) {
        int grow = row0 + i + 8 * half;
        if (grow < n) out[(size_t)grow * DF + wv * 16 + r] = acc[i];
    }
}

// ---------------------------------------------------------------------------
// Degree / normalization
// ---------------------------------------------------------------------------
__global__ void k_init_deg(float* deg, int n) {
    int i = blockIdx.x * blockDim.x + threadIdx.x;
    if (i < n) deg[i] = 1.0f;                 // self loop
}

__global__ void k_deg_edges(const int* __restrict__ dst, float* deg, int e) {
    int i = blockIdx.x * blockDim.x + threadIdx.x;
    if (i < e) unsafeAtomicAdd(&deg[dst[i]], 1.0f);
}

__global__ void k_dinv(float* deg, int n) {
    int i = blockIdx.x * blockDim.x + threadIdx.x;
    if (i < n) deg[i] = rsqrtf(fmaxf(deg[i], 1.0f));
}

// ---------------------------------------------------------------------------
// out[node] = h[node] * dinv[node]^2 + bias   (self-loop term + bias, fused)
// one float4 per thread
// ---------------------------------------------------------------------------
__global__ void k_agg_init(const float* __restrict__ h, const float* __restrict__ dinv,
                           const float* __restrict__ bias, float* __restrict__ out, int n)
{
    int i = blockIdx.x * blockDim.x + threadIdx.x;   // over n*32 float4s
    if (i >= n * 32) return;
    int node = i >> 5;
    int c4   = (i & 31) * 4;
    float s  = dinv[node]; s = s * s;
    float4 v = *(const float4*)(h + (size_t)node * DF + c4);
    float4 b = *(const float4*)(bias + c4);
    float4 o;
    o.x = fmaf(v.x, s, b.x); o.y = fmaf(v.y, s, b.y);
    o.z = fmaf(v.z, s, b.z); o.w = fmaf(v.w, s, b.w);
    *(float4*)(out + (size_t)node * DF + c4) = o;
}

// ---------------------------------------------------------------------------
// Edge scatter: one wave per edge, float4 per lane, native f32 atomics.
// h[src] stays L2-resident (51 MB < 192 MB L2).
// ---------------------------------------------------------------------------
__global__ __launch_bounds__(256) void k_agg_edges(
    const float* __restrict__ h, const int* __restrict__ src,
    const int* __restrict__ dst, const float* __restrict__ dinv,
    float* __restrict__ out, int e)
{
    int ed   = blockIdx.x * 8 + (threadIdx.x >> 5);
    int lane = threadIdx.x & 31;
    if (ed >= e) return;
    int s = src[ed];
    int d = dst[ed];
    float norm = dinv[s] * dinv[d];
    float4 v = *(const float4*)(h + (size_t)s * DF + lane * 4);
    float* o = out + (size_t)d * DF + lane * 4;
    unsafeAtomicAdd(o + 0, v.x * norm);
    unsafeAtomicAdd(o + 1, v.y * norm);
    unsafeAtomicAdd(o + 2, v.z * norm);
    unsafeAtomicAdd(o + 3, v.w * norm);
}

// ---------------------------------------------------------------------------
// Final FC: logits = relu(h) @ Wfc + bfc, C=2. One wave32 per node.
// ---------------------------------------------------------------------------
__global__ __launch_bounds__(256) void k_fc(
    const float* __restrict__ h, const float* __restrict__ Wfc,
    const float* __restrict__ bfc, float* __restrict__ out, int n)
{
    int node = (blockIdx.x * 256 + threadIdx.x) >> 5;
    int lane = threadIdx.x & 31;
    if (node >= n) return;
    const float* row = h + (size_t)node * DF;
    float a0 = 0.f, a1 = 0.f;
    #pragma unroll
    for (int d = lane; d < DF; d += 32) {
        float v = fmaxf(row[d], 0.f);
        a0 = fmaf(v, Wfc[d * 2 + 0], a0);
        a1 = fmaf(v, Wfc[d * 2 + 1], a1);
    }
    #pragma unroll
    for (int off = 16; off > 0; off >>= 1) {
        a0 += __shfl_xor(a0, off, 32);
        a1 += __shfl_xor(a1, off, 32);
    }
    if (lane == 0) {
        out[(size_t)node * 2 + 0] = a0 + bfc[0];
        out[(size_t)node * 2 + 1] = a1 + bfc[1];
    }
}

// ---------------------------------------------------------------------------
extern "C" void kernel_launch(void* const* d_in, const int* in_sizes, int n_in,
                              void* d_out, int out_size, void* d_ws, size_t ws_size,
                              hipStream_t stream)
{
    const float* x   = (const float*)d_in[0];
    const int*   ei  = (const int*)d_in[1];
    const float* W1  = (const float*)d_in[2];
    const float* b1  = (const float*)d_in[3];
    const float* W2  = (const float*)d_in[4];
    const float* b2  = (const float*)d_in[5];
    const float* Wfc = (const float*)d_in[6];
    const float* bfc = (const float*)d_in[7];
    float* out = (float*)d_out;

    const int n = in_sizes[0] / DF;   // 100000
    const int e = in_sizes[1] / 2;    // 640000
    const int* src = ei;
    const int* dst = ei + e;

    float* buf0 = (float*)d_ws;                    // n*128 floats
    float* buf1 = buf0 + (size_t)n * DF;           // n*128 floats
    float* dinv = buf1 + (size_t)n * DF;           // n floats (deg -> dinv in place)

    dim3 b256(256);
    const int nb   = (n + 255) / 256;
    const int ebT  = (e + 255) / 256;
    const int ebW  = (e + 7) / 8;        // 8 waves (edges) per block
    const int gemB = (n + 15) / 16;
    const int v4B  = ((n * 32) + 255) / 256;

    // degrees -> dinv
    k_init_deg <<<nb,  b256, 0, stream>>>(dinv, n);
    k_deg_edges<<<ebT, b256, 0, stream>>>(dst, dinv, e);
    k_dinv     <<<nb,  b256, 0, stream>>>(dinv, n);

    // layer 1: h1 = agg(x @ W1) + b1   (ReLU deferred to layer-2 GEMM input)
    gcn_gemm_wmma<<<gemB, b256, 0, stream>>>(x, W1, buf0, n, 0);
    k_agg_init   <<<v4B,  b256, 0, stream>>>(buf0, dinv, b1, buf1, n);
    k_agg_edges  <<<ebW,  b256, 0, stream>>>(buf0, src, dst, dinv, buf1, e);

    // layer 2: h2 = agg(relu(h1) @ W2) + b2
    gcn_gemm_wmma<<<gemB, b256, 0, stream>>>(buf1, W2, buf0, n, 1);
    k_agg_init   <<<v4B,  b256, 0, stream>>>(buf0, dinv, b2, buf1, n);
    k_agg_edges  <<<ebW,  b256, 0, stream>>>(buf0, src, dst, dinv, buf1, e);

    // logits = relu(h2) @ Wfc + bfc
    k_fc<<<v4B, b256, 0, stream>>>(buf1, Wfc, bfc, out, n);
}